// ResonanceMapping_20280835572110
// MI455X (gfx1250) — compile-verified
//
#include <hip/hip_runtime.h>
#include <hip/hip_bf16.h>
#include <math.h>

// ---------------------------------------------------------------------------
// Hyperbolic attention forward, MI455X (gfx1250, wave32).
// All S x S work is driven by fp32 WMMA (v_wmma_f32_16x16x4_f32) Gram GEMMs;
// S x S intermediates (~25 MB) stay L2-resident (192 MB L2).
// B-operands of the long K-loop GEMMs are pre-transposed so every WMMA
// fragment fetch is a single 8-byte global_load_b64.
// ---------------------------------------------------------------------------

constexpr int NS  = 1024;          // sequence length S
constexpr int ND  = 64;            // head dim D
constexpr int NSS = NS * NS;

#define BND  (1.0f - 1e-7f)

typedef float v2f __attribute__((ext_vector_type(2)));
typedef float v8f __attribute__((ext_vector_type(8)));

static __device__ __forceinline__ v8f wmma4(v2f a, v2f b, v8f c) {
  // D = A(16x4 f32) * B(4x16 f32) + C(16x16 f32), exact fp32 tensor op.
  return __builtin_amdgcn_wmma_f32_16x16x4_f32(
      /*neg_a=*/false, a, /*neg_b=*/false, b,
      /*c_mod=*/(short)0, c, /*reuse_a=*/false, /*reuse_b=*/false);
}

static __device__ __forceinline__ v2f ld2(const float* p) {
  return *(const v2f*)p;           // 8-byte aligned global load
}

// ---------------------------------------------------------------------------
// out[s, o] = sum_k in[s, k] * W[o, k] + bias[o]   ([NS,ND] x [ND,ND]^T)
// One wave computes a 16-row x 64-col block; 8 waves per 256-thread block.
// ---------------------------------------------------------------------------
__global__ void proj_kernel(const float* __restrict__ in,
                            const float* __restrict__ Wt,
                            const float* __restrict__ bias,
                            float* __restrict__ out) {
  const int wave = threadIdx.x >> 5;
  const int lane = threadIdx.x & 31;
  const int half = lane >> 4, l16 = lane & 15;
  const int I0 = (blockIdx.x * 8 + wave) * 16;

  v8f acc[4] = {};
  for (int k0 = 0; k0 < ND; k0 += 4) {
    const int kr = k0 + 2 * half;
    v2f a = ld2(&in[(I0 + l16) * ND + kr]);
#pragma unroll
    for (int t = 0; t < 4; ++t) {
      v2f b = ld2(&Wt[(t * 16 + l16) * ND + kr]);   // B_op[k][o] = W[o][k]
      acc[t] = wmma4(a, b, acc[t]);
    }
  }
#pragma unroll
  for (int t = 0; t < 4; ++t) {
    const int col = t * 16 + l16;
    const float bv = bias[col];
#pragma unroll
    for (int g = 0; g < 8; ++g) {
      const int row = I0 + g + 8 * half;
      out[row * ND + col] = acc[t][g] + bv;
    }
  }
}

// ---------------------------------------------------------------------------
// LDS-tiled transpose [NS,ND] -> [ND,NS]; padded row (17) avoids bank
// conflicts on the 64-bank LDS.
// ---------------------------------------------------------------------------
__global__ void transpose_kernel(const float* __restrict__ in,
                                 float* __restrict__ out) {
  __shared__ float tile[16][17];
  const int bs = blockIdx.x * 16;         // row (s) tile base
  const int bd = blockIdx.y * 16;         // col (d) tile base
  const int tx = threadIdx.x & 15, ty = threadIdx.x >> 4;
  tile[ty][tx] = in[(bs + ty) * ND + bd + tx];
  __syncthreads();
  out[(bd + ty) * NS + bs + tx] = tile[tx][ty];
}

// ---------------------------------------------------------------------------
// expmap0 per row: out = tanh(n)/n * in, n = sqrt(sum in^2 + 1e-15);
// also emits squared norm of the mapped point.
// ---------------------------------------------------------------------------
__global__ void expmap_kernel(const float* __restrict__ in,
                              float* __restrict__ out,
                              float* __restrict__ n2out) {
  const int row = blockIdx.x * blockDim.x + threadIdx.x;
  if (row >= NS) return;
  const float* p = in + row * ND;
  float s = 0.f;
#pragma unroll 8
  for (int d = 0; d < ND; ++d) s += p[d] * p[d];
  const float n = sqrtf(s + 1e-15f);
  const float t = tanhf(n) / n;
  float* o = out + row * ND;
#pragma unroll 8
  for (int d = 0; d < ND; ++d) o[d] = t * p[d];
  n2out[row] = t * t * s;
}

// ---------------------------------------------------------------------------
// Gram tile (Qh * Kh^T) with fused geodesic-distance epilogue.
// dist = 2 artanh( min(BND, sqrt(num2)/den) ) from only (xy, x2, y2).
// ---------------------------------------------------------------------------
__global__ void dist_kernel(const float* __restrict__ Qh,
                            const float* __restrict__ Kh,
                            const float* __restrict__ qn2,
                            const float* __restrict__ kn2,
                            float* __restrict__ Dist) {
  const int wave = threadIdx.x >> 5;
  const int lane = threadIdx.x & 31;
  const int half = lane >> 4, l16 = lane & 15;
  const int tile = blockIdx.x * 8 + wave;
  const int I0 = (tile >> 6) << 4;
  const int J0 = (tile & 63) << 4;

  v8f acc = {};
  for (int k0 = 0; k0 < ND; k0 += 4) {
    const int kr = k0 + 2 * half;
    v2f a = ld2(&Qh[(I0 + l16) * ND + kr]);
    v2f b = ld2(&Kh[(J0 + l16) * ND + kr]);
    acc = wmma4(a, b, acc);
  }
  const int col = J0 + l16;
  const float y2 = kn2[col];
#pragma unroll
  for (int g = 0; g < 8; ++g) {
    const int row = I0 + g + 8 * half;
    const float x2 = qn2[row];
    const float xy = acc[g];
    const float aa = 1.f - 2.f * xy + y2;
    const float bb = 1.f - x2;
    const float den = fmaxf(1.f - 2.f * xy + x2 * y2, 1e-15f);
    const float num2 = aa * aa * x2 - 2.f * aa * bb * xy + bb * bb * y2;
    const float nn = sqrtf(num2 / (den * den) + 1e-15f);
    Dist[row * NS + col] = 2.f * atanhf(fminf(nn, BND));
  }
}

// ---------------------------------------------------------------------------
// Per-row: min -> w_un = exp(-(d-min)) -> w_sum(+1e-8) -> entropy/eff_mass.
// Fixed-order tree reductions (bitwise deterministic).
// ---------------------------------------------------------------------------
__global__ void softrow_kernel(const float* __restrict__ Dist,
                               float* __restrict__ Wun,
                               float* __restrict__ wsum,
                               float* __restrict__ effm) {
  const int row = blockIdx.x, tid = threadIdx.x;
  __shared__ float red[256];
  const float* dr = Dist + (size_t)row * NS;
  float* wr = Wun + (size_t)row * NS;

  float m = 3.4e38f;
  for (int j = tid; j < NS; j += 256) m = fminf(m, dr[j]);
  red[tid] = m; __syncthreads();
  for (int s = 128; s > 0; s >>= 1) { if (tid < s) red[tid] = fminf(red[tid], red[tid + s]); __syncthreads(); }
  m = red[0]; __syncthreads();

  float sum = 0.f;
  for (int j = tid; j < NS; j += 256) { float w = expf(-(dr[j] - m)); wr[j] = w; sum += w; }
  red[tid] = sum; __syncthreads();
  for (int s = 128; s > 0; s >>= 1) { if (tid < s) red[tid] += red[tid + s]; __syncthreads(); }
  const float ws = red[0] + 1e-8f; __syncthreads();

  const float inv = 1.f / ws;
  float e = 0.f;
  for (int j = tid; j < NS; j += 256) { float p = wr[j] * inv; e -= p * logf(p + 1e-8f); }
  red[tid] = e; __syncthreads();
  for (int s = 128; s > 0; s >>= 1) { if (tid < s) red[tid] += red[tid + s]; __syncthreads(); }
  if (tid == 0) { wsum[row] = ws; effm[row] = expf(red[0]); }
}

// ---------------------------------------------------------------------------
// Fused h = P @ V and centroid_euc = P @ K, with P = w_un / w_sum scaled
// into the A fragment (row scaling). B operands come from pre-transposed
// Vt/Kt [ND,NS] so each fragment is one 8-byte load. One wave: 16x64 of both.
// ---------------------------------------------------------------------------
__global__ void pv_kernel(const float* __restrict__ Wun,
                          const float* __restrict__ wsum,
                          const float* __restrict__ Vt,
                          const float* __restrict__ Kt,
                          float* __restrict__ H,
                          float* __restrict__ Ce) {
  const int wave = threadIdx.x >> 5;
  const int lane = threadIdx.x & 31;
  const int half = lane >> 4, l16 = lane & 15;
  const int I0 = (blockIdx.x * 8 + wave) * 16;
  const float invw = 1.f / wsum[I0 + l16];   // A-fragment row is I0+l16
  const float* arow = Wun + (size_t)(I0 + l16) * NS;

  v8f ah[4] = {}, ac[4] = {};
  for (int k0 = 0; k0 < NS; k0 += 4) {
    const int kr = k0 + 2 * half;
    if ((k0 & 63) == 0 && k0 + 256 < NS)
      __builtin_prefetch(arow + k0 + 256, 0, 1);   // global_prefetch_b8
    v2f a = ld2(arow + kr);
    a.x *= invw; a.y *= invw;
#pragma unroll
    for (int t = 0; t < 4; ++t) {
      const int col = t * 16 + l16;
      v2f bv = ld2(&Vt[(size_t)col * NS + kr]);
      v2f bk = ld2(&Kt[(size_t)col * NS + kr]);
      ah[t] = wmma4(a, bv, ah[t]);
      ac[t] = wmma4(a, bk, ac[t]);
    }
  }
#pragma unroll
  for (int t = 0; t < 4; ++t) {
    const int col = t * 16 + l16;
#pragma unroll
    for (int g = 0; g < 8; ++g) {
      const int row = I0 + g + 8 * half;
      H[row * ND + col]  = ah[t][g];
      Ce[row * ND + col] = ac[t][g];
    }
  }
}

// ---------------------------------------------------------------------------
// Gram (Ch * Kh^T) with fused epilogue producing, per (q,j):
//   VC = wnv * c2k^2   (variance integrand)
//   AL = sqrt(w+1e-8) * scal * (-a)    coeff of centroid c_q in weighted_k
//   BE = sqrt(w+1e-8) * scal * b       coeff of key k_j
// where k_local = max(b,eps) * artanh(n)/n * (-a c + b k)/den.
// ---------------------------------------------------------------------------
__global__ void geo_kernel(const float* __restrict__ Ch,
                           const float* __restrict__ Kh,
                           const float* __restrict__ cn2,
                           const float* __restrict__ kn2,
                           const float* __restrict__ Wun,
                           const float* __restrict__ wsum,
                           float* __restrict__ AL,
                           float* __restrict__ BE,
                           float* __restrict__ VC) {
  const int wave = threadIdx.x >> 5;
  const int lane = threadIdx.x & 31;
  const int half = lane >> 4, l16 = lane & 15;
  const int tile = blockIdx.x * 8 + wave;
  const int I0 = (tile >> 6) << 4;
  const int J0 = (tile & 63) << 4;

  v8f acc = {};
  for (int k0 = 0; k0 < ND; k0 += 4) {
    const int kr = k0 + 2 * half;
    v2f a = ld2(&Ch[(I0 + l16) * ND + kr]);
    v2f b = ld2(&Kh[(J0 + l16) * ND + kr]);
    acc = wmma4(a, b, acc);
  }
  const int col = J0 + l16;
  const float y2 = kn2[col];
#pragma unroll
  for (int g = 0; g < 8; ++g) {
    const int row = I0 + g + 8 * half;
    const float x2 = cn2[row];
    const float xy = acc[g];
    const float aa = 1.f - 2.f * xy + y2;
    const float bb = 1.f - x2;
    const float den = fmaxf(1.f - 2.f * xy + x2 * y2, 1e-15f);
    const float num2 = aa * aa * x2 - 2.f * aa * bb * xy + bb * bb * y2;
    const float nn = sqrtf(num2 / (den * den) + 1e-15f);
    const float tt = atanhf(fminf(nn, BND));
    const float c2k = 2.f * tt;
    const int idx = row * NS + col;
    const float w = Wun[idx];
    const float wnv = w / wsum[row];
    VC[idx] = wnv * c2k * c2k;
    const float scal = fmaxf(bb, 1e-15f) * (tt / nn) / den;
    const float sw = sqrtf(w + 1e-8f);
    AL[idx] = -sw * scal * aa;
    BE[idx] =  sw * scal * bb;
  }
}

// Deterministic per-row sum over an S x S field.
__global__ void rowsum_kernel(const float* __restrict__ M, float* __restrict__ out) {
  const int row = blockIdx.x, tid = threadIdx.x;
  __shared__ float red[256];
  const float* mr = M + (size_t)row * NS;
  float s = 0.f;
  for (int j = tid; j < NS; j += 256) s += mr[j];
  red[tid] = s; __syncthreads();
  for (int k = 128; k > 0; k >>= 1) { if (tid < k) red[tid] += red[tid + k]; __syncthreads(); }
  if (tid == 0) out[row] = red[0];
}

// ---------------------------------------------------------------------------
// Power-iteration prep: v_r = normalize(mode? r1*Ch + v2 : v_init), and
// s1 = Ch . v_r (per row).
// ---------------------------------------------------------------------------
__global__ void prep_kernel(const float* __restrict__ vinit,
                            const float* __restrict__ v2,
                            const float* __restrict__ r1,
                            const float* __restrict__ Ch,
                            float* __restrict__ vr,
                            float* __restrict__ s1, int mode) {
  const int row = blockIdx.x * blockDim.x + threadIdx.x;
  if (row >= NS) return;
  float v[ND];
  if (mode == 0) {
#pragma unroll 8
    for (int d = 0; d < ND; ++d) v[d] = vinit[row * ND + d];
  } else {
    const float r = r1[row];
#pragma unroll 8
    for (int d = 0; d < ND; ++d) v[d] = r * Ch[row * ND + d] + v2[row * ND + d];
  }
  float n2 = 0.f;
#pragma unroll 8
  for (int d = 0; d < ND; ++d) n2 += v[d] * v[d];
  const float inv = 1.f / fmaxf(sqrtf(n2), 1e-12f);
  float s = 0.f;
#pragma unroll 8
  for (int d = 0; d < ND; ++d) {
    const float x = v[d] * inv;
    vr[row * ND + d] = x;
    s += Ch[row * ND + d] * x;
  }
  s1[row] = s;
}

// ---------------------------------------------------------------------------
// T = v_r * Kh^T tile, fused:  proj = AL*s1 + BE*T;  M = proj*BE;  PA = proj*AL.
// ---------------------------------------------------------------------------
__global__ void titer_kernel(const float* __restrict__ vr,
                             const float* __restrict__ Kh,
                             const float* __restrict__ AL,
                             const float* __restrict__ BE,
                             const float* __restrict__ s1,
                             float* __restrict__ M,
                             float* __restrict__ PA) {
  const int wave = threadIdx.x >> 5;
  const int lane = threadIdx.x & 31;
  const int half = lane >> 4, l16 = lane & 15;
  const int tile = blockIdx.x * 8 + wave;
  const int I0 = (tile >> 6) << 4;
  const int J0 = (tile & 63) << 4;

  v8f acc = {};
  for (int k0 = 0; k0 < ND; k0 += 4) {
    const int kr = k0 + 2 * half;
    v2f a = ld2(&vr[(I0 + l16) * ND + kr]);
    v2f b = ld2(&Kh[(J0 + l16) * ND + kr]);
    acc = wmma4(a, b, acc);
  }
  const int col = J0 + l16;
#pragma unroll
  for (int g = 0; g < 8; ++g) {
    const int row = I0 + g + 8 * half;
    const int idx = row * NS + col;
    const float al = AL[idx], be = BE[idx];
    const float proj = al * s1[row] + be * acc[g];
    M[idx]  = proj * be;
    PA[idx] = proj * al;
  }
}

// v2 = M @ Kh  ([NS,NS] x [NS,ND]); B from pre-transposed KhT [ND,NS].
__global__ void miter_kernel(const float* __restrict__ M,
                             const float* __restrict__ KhT,
                             float* __restrict__ v2out) {
  const int wave = threadIdx.x >> 5;
  const int lane = threadIdx.x & 31;
  const int half = lane >> 4, l16 = lane & 15;
  const int I0 = (blockIdx.x * 8 + wave) * 16;
  const float* arow = M + (size_t)(I0 + l16) * NS;

  v8f acc[4] = {};
  for (int k0 = 0; k0 < NS; k0 += 4) {
    const int kr = k0 + 2 * half;
    if ((k0 & 63) == 0 && k0 + 256 < NS)
      __builtin_prefetch(arow + k0 + 256, 0, 1);   // global_prefetch_b8
    v2f a = ld2(arow + kr);
#pragma unroll
    for (int t = 0; t < 4; ++t) {
      const int col = t * 16 + l16;
      v2f b = ld2(&KhT[(size_t)col * NS + kr]);
      acc[t] = wmma4(a, b, acc[t]);
    }
  }
#pragma unroll
  for (int t = 0; t < 4; ++t) {
    const int col = t * 16 + l16;
#pragma unroll
    for (int g = 0; g < 8; ++g)
      v2out[(I0 + g + 8 * half) * ND + col] = acc[t][g];
  }
}

// ---------------------------------------------------------------------------
// Per-row bifurcation epilogue: parallel transport / fallback direction,
// asinh compression, Gumbel straight-through pitchfork, tangent rescale,
// exp/log map round trip, gated fuse.
// ---------------------------------------------------------------------------
__global__ void final_kernel(const float* __restrict__ H,
                             const float* __restrict__ Qh,
                             const float* __restrict__ Ch,
                             const float* __restrict__ cn2,
                             const float* __restrict__ Vr,
                             const float* __restrict__ var,
                             const float* __restrict__ effm,
                             const float* __restrict__ tau,
                             const float* __restrict__ gamma,
                             const float* __restrict__ tscale,
                             const float* __restrict__ gu,
                             float* __restrict__ HF) {
  const int row = blockIdx.x * blockDim.x + threadIdx.x;
  if (row >= NS) return;
  const float* h  = H  + row * ND;
  const float* qh = Qh + row * ND;
  const float* ch = Ch + row * ND;
  const float* vr = Vr + row * ND;

  const float variance = var[row];
  const bool useg = variance > 1e-5f;
  float invden = 0.f, aq = 0.f, ac = 0.f, invnf = 0.f;
  if (useg) {
    invden = 1.f / fmaxf(1.f - cn2[row], 1e-15f);
  } else {
    float q2 = 0.f, c2 = 0.f;
    for (int d = 0; d < ND; ++d) { q2 += qh[d] * qh[d]; c2 += ch[d] * ch[d]; }
    const float nq = sqrtf(q2 + 1e-15f), nc = sqrtf(c2 + 1e-15f);
    aq = atanhf(fminf(nq, BND)) / nq;
    ac = atanhf(fminf(nc, BND)) / nc;
    float f2 = 0.f;
    for (int d = 0; d < ND; ++d) { float f = aq * qh[d] - ac * ch[d]; f2 += f * f; }
    invnf = 1.f / fmaxf(sqrtf(f2), 1e-8f);
  }
  auto wpf = [&](int d) -> float {
    return useg ? vr[d] * invden : (aq * qh[d] - ac * ch[d]) * invnf;
  };

  float h2 = 0.f;
  for (int d = 0; d < ND; ++d) h2 += h[d] * h[d];
  const float hn  = sqrtf(h2 + 1e-15f);
  const float asc = asinhf(hn) / (hn + 1e-8f);

  float x = 0.f, hc2 = 0.f;
  for (int d = 0; d < ND; ++d) {
    const float hcd = asc * h[d];
    x += hcd * wpf(d); hc2 += hcd * hcd;
  }
  const float hmag = sqrtf(hc2 + 1e-15f);
  const float tension = variance - tau[0] * effm[row];
  const float amp = hmag * tanhf(tension) * (tension > 0.f ? 1.f : 0.f);

  const float g0 = -logf(-logf(gu[2 * row + 0]));
  const float g1 = -logf(-logf(gu[2 * row + 1]));
  const float epsg = (g0 >= g1) ? 1.f : -1.f;   // straight-through hard value
  const float dc = tanhf(epsg * amp - x);

  float hp2 = 0.f;
  for (int d = 0; d < ND; ++d) { float t = asc * h[d] + dc * wpf(d); hp2 += t * t; }
  const float hpn = sqrtf(hp2 + 1e-15f) + 1e-8f;
  const float sc1 = tanhf(hpn * tscale[0]) / hpn;
  const float pb2 = sc1 * sc1 * hp2;
  const float pbn = sqrtf(pb2 + 1e-15f);
  const float sc2 = tanhf(pbn) / pbn;
  const float hy2 = sc2 * sc2 * pb2;
  const float nh  = sqrtf(hy2 + 1e-15f);
  const float sc3 = atanhf(fminf(nh, BND)) / nh;
  const float gate = 1.f / (1.f + expf(-gamma[0]));
  const float cb = sc3 * sc2 * sc1;

  float* o = HF + row * ND;
  for (int d = 0; d < ND; ++d) {
    const float hpd = asc * h[d] + dc * wpf(d);
    o[d] = (1.f - gate) * h[d] + gate * cb * hpd;
  }
}

// ---------------------------------------------------------------------------
extern "C" void kernel_launch(void* const* d_in, const int* in_sizes, int n_in,
                              void* d_out, int out_size, void* d_ws, size_t ws_size,
                              hipStream_t stream) {
  const float* q_in  = (const float*)d_in[0];
  const float* k_in  = (const float*)d_in[1];
  const float* v_in  = (const float*)d_in[2];
  const float* Wq    = (const float*)d_in[3];
  const float* bq    = (const float*)d_in[4];
  const float* Wk    = (const float*)d_in[5];
  const float* bk    = (const float*)d_in[6];
  const float* Wv    = (const float*)d_in[7];
  const float* bv    = (const float*)d_in[8];
  const float* Wo    = (const float*)d_in[9];
  const float* bo    = (const float*)d_in[10];
  const float* tau   = (const float*)d_in[11];
  const float* gamma = (const float*)d_in[12];
  const float* tsc   = (const float*)d_in[13];
  const float* gu    = (const float*)d_in[14];
  const float* vri   = (const float*)d_in[15];
  float* out = (float*)d_out;

  const int SD = NS * ND;
  float* ws  = (float*)d_ws;
  float* Q   = ws + 0 * SD;
  float* Kp  = ws + 1 * SD;
  float* Vp  = ws + 2 * SD;
  float* Qh  = ws + 3 * SD;
  float* Kh  = ws + 4 * SD;
  float* Ce  = ws + 5 * SD;
  float* Ch  = ws + 6 * SD;
  float* H   = ws + 7 * SD;
  float* Vr  = ws + 8 * SD;
  float* V2  = ws + 9 * SD;
  float* HF  = ws + 10 * SD;
  float* Vt  = ws + 11 * SD;   // V^T   [ND,NS]
  float* Kt  = ws + 12 * SD;   // K^T   [ND,NS]
  float* KhT = ws + 13 * SD;   // Kh^T  [ND,NS]
  float* qn2 = ws + 14 * SD;
  float* kn2 = qn2 + NS;
  float* cn2 = kn2 + NS;
  float* wsm = cn2 + NS;
  float* efm = wsm + NS;
  float* var = efm + NS;
  float* r1  = var + NS;
  float* s1  = r1 + NS;
  float* Wun = ws + 14 * SD + 8 * NS;
  float* AL  = Wun + (size_t)NSS;
  float* BE  = AL  + (size_t)NSS;
  float* X1  = BE  + (size_t)NSS;   // dist -> variance integrand -> M
  float* X2  = X1  + (size_t)NSS;   // PA scratch

  const dim3 tg(NS / 16, ND / 16);

  // Projections (WMMA): Q/K/V = in @ W^T + b
  proj_kernel<<<8, 256, 0, stream>>>(q_in, Wq, bq, Q);
  proj_kernel<<<8, 256, 0, stream>>>(k_in, Wk, bk, Kp);
  proj_kernel<<<8, 256, 0, stream>>>(v_in, Wv, bv, Vp);
  expmap_kernel<<<4, 256, 0, stream>>>(Q,  Qh, qn2);
  expmap_kernel<<<4, 256, 0, stream>>>(Kp, Kh, kn2);
  transpose_kernel<<<tg, 256, 0, stream>>>(Vp, Vt);
  transpose_kernel<<<tg, 256, 0, stream>>>(Kp, Kt);
  transpose_kernel<<<tg, 256, 0, stream>>>(Kh, KhT);

  // All-pairs geodesic distance via Gram WMMA, then row softmax/entropy.
  dist_kernel<<<512, 256, 0, stream>>>(Qh, Kh, qn2, kn2, X1);
  softrow_kernel<<<NS, 256, 0, stream>>>(X1, Wun, wsm, efm);

  // h = P@V, centroid_euc = P@K (fused WMMA), map centroid to manifold.
  pv_kernel<<<8, 256, 0, stream>>>(Wun, wsm, Vt, Kt, H, Ce);
  expmap_kernel<<<4, 256, 0, stream>>>(Ce, Ch, cn2);

  // Centroid->key Gram + rank-2 tangent coefficients + variance integrand.
  geo_kernel<<<512, 256, 0, stream>>>(Ch, Kh, cn2, kn2, Wun, wsm, AL, BE, X1);
  rowsum_kernel<<<NS, 256, 0, stream>>>(X1, var);

  // Matrix-free TPCA power iteration (3 steps), all GEMMs on WMMA.
  prep_kernel<<<4, 256, 0, stream>>>(vri, V2, r1, Ch, Vr, s1, 0);
  for (int it = 0; it < 3; ++it) {
    titer_kernel<<<512, 256, 0, stream>>>(Vr, Kh, AL, BE, s1, X1, X2);
    rowsum_kernel<<<NS, 256, 0, stream>>>(X2, r1);
    miter_kernel<<<8, 256, 0, stream>>>(X1, KhT, V2);
    prep_kernel<<<4, 256, 0, stream>>>(vri, V2, r1, Ch, Vr, s1, 1);
  }

  // Bifurcation epilogue + gated fuse, then output projection (WMMA).
  final_kernel<<<4, 256, 0, stream>>>(H, Qh, Ch, cn2, Vr, var, efm,
                                      tau, gamma, tsc, gu, HF);
  proj_kernel<<<8, 256, 0, stream>>>(HF, Wo, bo, out);
}